// LinearAttention_11802570129790
// MI455X (gfx1250) — compile-verified
//
#include <hip/hip_runtime.h>

typedef __attribute__((ext_vector_type(2))) float v2f;
typedef __attribute__((ext_vector_type(8))) float v8f;
typedef __attribute__((ext_vector_type(4))) unsigned int v4u;
typedef __attribute__((ext_vector_type(8))) int v8i;
typedef __attribute__((ext_vector_type(4))) int v4i;

#define BDIM 4
#define LDIM 4096
#define HDIM 16
#define DDIM 128
#define EPSV 1e-6f
#define LSPLIT 8      // split-K parts for the KV GEMM
#define LCHUNK 32     // l-chunk staged in LDS per iteration
#define QSPLIT 32     // L-split for the q_global reduction
#define PSTR (DDIM + 16)  // LDS row stride; TDM pad (16 dw per 128 dw) produces this
#define ASTR (DDIM + 4)   // padded LDS row stride (out kernel)

// ---------------------------------------------------------------- zero scratch
__global__ void rala_zero_kernel(float* __restrict__ p, int n) {
  int i = blockIdx.x * 256 + threadIdx.x;
  if (i < n) p[i] = 0.0f;
}

// ------------------------------------------------- q_global = mean(Q)/sqrt(2048)
__global__ __launch_bounds__(128) void rala_qglobal_kernel(const float* __restrict__ Q,
                                                           float* __restrict__ qg) {
  int blk  = blockIdx.x;
  int part = blk % QSPLIT;
  int bh   = blk / QSPLIT;
  int b = bh / HDIM, h = bh % HDIM;
  int d = threadIdx.x;
  const int lspan = LDIM / QSPLIT;
  const float* base = Q + (((size_t)b * LDIM + (size_t)part * lspan) * HDIM + h) * DDIM + d;
  float s = 0.0f;
  for (int l = 0; l < lspan; ++l) s += base[(size_t)l * HDIM * DDIM];
  s *= (1.0f / (float)LDIM) * 0.022097086912079608f;  // 1/sqrt(2048)
  atomicAdd(&qg[bh * DDIM + d], s);
}

// --------------------------------------- alpha = softmax_l(q_global . K) * L
__global__ __launch_bounds__(256) void rala_alpha_kernel(const float* __restrict__ K,
                                                         const float* __restrict__ qg,
                                                         float* __restrict__ alpha) {
  __shared__ float qs[DDIM];
  __shared__ float red[256];
  int bh = blockIdx.x;
  int b = bh / HDIM, h = bh % HDIM;
  int tid = threadIdx.x;
  if (tid < DDIM) qs[tid] = qg[bh * DDIM + tid];
  __syncthreads();
  float* sc = alpha + (size_t)bh * LDIM;
  float lmax = -1e30f;
  for (int l = tid; l < LDIM; l += 256) {
    const float4* kp = (const float4*)(K + (((size_t)b * LDIM + l) * HDIM + h) * DDIM);
    float s = 0.0f;
    #pragma unroll
    for (int i = 0; i < DDIM / 4; ++i) {
      float4 k4 = kp[i];
      s += k4.x * qs[4 * i] + k4.y * qs[4 * i + 1] + k4.z * qs[4 * i + 2] + k4.w * qs[4 * i + 3];
    }
    sc[l] = s;
    lmax = fmaxf(lmax, s);
  }
  red[tid] = lmax; __syncthreads();
  for (int off = 128; off > 0; off >>= 1) {
    if (tid < off) red[tid] = fmaxf(red[tid], red[tid + off]);
    __syncthreads();
  }
  float gmax = red[0]; __syncthreads();
  float lsum = 0.0f;
  for (int l = tid; l < LDIM; l += 256) lsum += __expf(sc[l] - gmax);
  red[tid] = lsum; __syncthreads();
  for (int off = 128; off > 0; off >>= 1) {
    if (tid < off) red[tid] += red[tid + off];
    __syncthreads();
  }
  float scale = (float)LDIM / red[0];
  for (int l = tid; l < LDIM; l += 256) sc[l] = __expf(sc[l] - gmax) * scale;
}

// ---------------- Tensor Data Mover: 2D tile (rows x cols f32) -> LDS, padded rows
__device__ __forceinline__ unsigned lds_off(const void* p) {
  return (unsigned)(size_t)p;   // low 32 bits of generic shared pointer = LDS byte offset
}

__device__ __forceinline__ void tdm_load_tile_f32(unsigned lds_addr, const float* gptr,
                                                  unsigned rows, unsigned cols,
                                                  unsigned row_stride_elems) {
  unsigned long long ga = (unsigned long long)(size_t)gptr;
  v4u g0;
  g0.x = 1u;                                          // count=1, user descriptor
  g0.y = lds_addr;                                    // LDS byte address
  g0.z = (unsigned)(ga & 0xFFFFFFFFull);              // global_addr[31:0]
  g0.w = (unsigned)((ga >> 32) & 0x01FFFFFFull) | (2u << 30);  // global_addr[56:32] | type=2
  v8i g1;
  // data_size=4B (2), pad_enable, pad_interval=128 dwords (6), pad_amount=16 dwords (15)
  g1[0] = (int)((2u << 16) | (1u << 20) | (6u << 22) | (15u << 25));
  g1[1] = (int)(cols << 16);                          // tensor_dim0[15:0] at bits 63:48
  g1[2] = (int)((cols >> 16) | (rows << 16));         // tensor_dim0[31:16] | tensor_dim1[15:0]
  g1[3] = (int)((rows >> 16) | (cols << 16));         // tensor_dim1[31:16] | tile_dim0
  g1[4] = (int)rows;                                  // tile_dim1 (tile_dim2 = 0 -> 2D)
  g1[5] = (int)row_stride_elems;                      // tensor_dim0_stride[31:0]
  g1[6] = 0;
  g1[7] = 0;
  v4i z4 = {};
  v8i z8 = {};
  __builtin_amdgcn_tensor_load_to_lds(g0, g1, z4, z4, z8, 0);  // 6-arg toolchain form
}

// ---------- KV[d,e] = sum_l (phiK*alpha)[l,d] * V[l,e]; Ksum[d] = sum_l (phiK*alpha)[l,d]
// Double-buffered TDM staging overlapped with f32 WMMA.
__global__ __launch_bounds__(256) void rala_kv_kernel(const float* __restrict__ phiK,
                                                      const float* __restrict__ V,
                                                      const float* __restrict__ alpha,
                                                      float* __restrict__ KV,
                                                      float* __restrict__ Ksum) {
  __shared__ float P2[2][LCHUNK * PSTR];   // raw phiK tiles, [l][d], TDM-padded rows
  __shared__ float V2[2][LCHUNK * PSTR];   // V tiles, [l][e]
  __shared__ float As[2][LCHUNK];          // alpha for the staged l-chunk
  int blk  = blockIdx.x;
  int part = blk % LSPLIT;
  int bh   = blk / LSPLIT;
  int b = bh / HDIM, h = bh % HDIM;
  int tid  = threadIdx.x;
  int wave = tid >> 5, lane = tid & 31;
  int m    = lane & 15;
  int koff = (lane >> 4) << 1;  // lanes 0-15 -> K=0,1 ; lanes 16-31 -> K=2,3
  int lbase = part * (LDIM / LSPLIT);
  const float* aptr = alpha + (size_t)bh * LDIM;
  const int nch = (LDIM / LSPLIT) / LCHUNK;

  v8f acc[8];
  #pragma unroll
  for (int e = 0; e < 8; ++e) { v8f z = {}; acc[e] = z; }
  float ks = 0.0f;

  // element offset of (b, l, h, 0); l -> l+1 stride is HDIM*DDIM elements
  size_t base0 = ((size_t)b * LDIM + lbase) * HDIM * DDIM + (size_t)h * DDIM;
  const size_t lstride = (size_t)HDIM * DDIM;

  if (wave == 0) {  // prologue: DMA chunk 0 into buffer 0
    tdm_load_tile_f32(lds_off(&P2[0][0]), phiK + base0, LCHUNK, DDIM, HDIM * DDIM);
    tdm_load_tile_f32(lds_off(&V2[0][0]), V + base0, LCHUNK, DDIM, HDIM * DDIM);
  }

  for (int ch = 0; ch < nch; ++ch) {
    int buf = ch & 1;
    if (wave == 0) {
      if (ch + 1 < nch) {  // prefetch next chunk into the other buffer
        size_t nb = base0 + (size_t)(ch + 1) * LCHUNK * lstride;
        tdm_load_tile_f32(lds_off(&P2[buf ^ 1][0]), phiK + nb, LCHUNK, DDIM, HDIM * DDIM);
        tdm_load_tile_f32(lds_off(&V2[buf ^ 1][0]), V + nb, LCHUNK, DDIM, HDIM * DDIM);
        __builtin_amdgcn_s_wait_tensorcnt(2);  // TDM in-order: chunk ch is complete
      } else {
        __builtin_amdgcn_s_wait_tensorcnt(0);
      }
    }
    if (tid < LCHUNK) As[buf][tid] = aptr[lbase + ch * LCHUNK + tid];
    __syncthreads();

    if (tid < DDIM) {   // Ksum partial: sum_l phiK[l,d]*alpha[l]
      #pragma unroll
      for (int lr = 0; lr < LCHUNK; ++lr) ks += P2[buf][lr * PSTR + tid] * As[buf][lr];
    }
    // A = (alpha*phiK)^T (16 d-rows per wave x K=4 l), B = V (K=4 l x 16 e)
    #pragma unroll
    for (int kk = 0; kk < LCHUNK; kk += 4) {
      float a0s = As[buf][kk + koff];
      float a1s = As[buf][kk + koff + 1];
      v2f a;
      a.x = P2[buf][(kk + koff) * PSTR + 16 * wave + m] * a0s;
      a.y = P2[buf][(kk + koff + 1) * PSTR + 16 * wave + m] * a1s;
      #pragma unroll
      for (int e = 0; e < 8; ++e) {
        v2f bb;
        bb.x = V2[buf][(kk + koff) * PSTR + 16 * e + m];
        bb.y = V2[buf][(kk + koff + 1) * PSTR + 16 * e + m];
        acc[e] = __builtin_amdgcn_wmma_f32_16x16x4_f32(false, a, false, bb,
                                                       (short)0, acc[e], false, false);
      }
    }
    __syncthreads();  // all waves done with buf before TDM overwrites it next iter
  }

  // merge split-K partials
  int rowoff = (lane >> 4) << 3;  // C/D layout: lanes 16-31 hold rows M+8
  float* kvb = KV + (size_t)bh * DDIM * DDIM;
  #pragma unroll
  for (int e = 0; e < 8; ++e) {
    #pragma unroll
    for (int r = 0; r < 8; ++r) {
      int d  = 16 * wave + r + rowoff;
      int ee = 16 * e + m;
      atomicAdd(&kvb[(size_t)d * DDIM + ee], acc[e][r]);
    }
  }
  if (tid < DDIM) atomicAdd(&Ksum[bh * DDIM + tid], ks);
}

// -------- out[l,e] = (phiQ[l,:] . KV[:,e]) / (phiQ[l,:] . Ksum + eps)
__global__ __launch_bounds__(256) void rala_out_kernel(const float* __restrict__ phiQ,
                                                       const float* __restrict__ KV,
                                                       const float* __restrict__ Ksum,
                                                       float* __restrict__ out) {
  __shared__ float Aq[16 * ASTR];
  __shared__ float kss[DDIM];
  __shared__ float den[16];
  int blk  = blockIdx.x;
  int lblk = blk % (LDIM / 16);
  int bh   = blk / (LDIM / 16);
  int b = bh / HDIM, h = bh % HDIM;
  int l0 = lblk * 16;
  int tid  = threadIdx.x;
  int wave = tid >> 5, lane = tid & 31;
  int m    = lane & 15;
  int koff = (lane >> 4) << 1;

  for (int i = tid; i < 16 * (DDIM / 4); i += 256) {
    int r  = i >> 5;
    int d4 = i & 31;
    float4 q4 = ((const float4*)(phiQ + (((size_t)b * LDIM + l0 + r) * HDIM + h) * DDIM))[d4];
    *(float4*)&Aq[r * ASTR + d4 * 4] = q4;
  }
  if (tid < DDIM) kss[tid] = Ksum[bh * DDIM + tid];
  __syncthreads();
  if (tid < 16) {
    float s = 0.0f;
    #pragma unroll 8
    for (int d = 0; d < DDIM; ++d) s += Aq[tid * ASTR + d] * kss[d];
    den[tid] = s + EPSV;
  }
  __syncthreads();

  v8f acc = {};
  const float* Bmat = KV + (size_t)bh * DDIM * DDIM + 16 * wave + m;  // this wave's e-column
  #pragma unroll 8
  for (int k0 = 0; k0 < DDIM; k0 += 4) {
    v2f a;
    a.x = Aq[m * ASTR + k0 + koff];
    a.y = Aq[m * ASTR + k0 + koff + 1];
    v2f bb;
    bb.x = Bmat[(size_t)(k0 + koff) * DDIM];
    bb.y = Bmat[(size_t)(k0 + koff + 1) * DDIM];
    acc = __builtin_amdgcn_wmma_f32_16x16x4_f32(false, a, false, bb,
                                                (short)0, acc, false, false);
  }

  int rowoff = (lane >> 4) << 3;
  int e = 16 * wave + m;
  #pragma unroll
  for (int r = 0; r < 8; ++r) {
    int row = r + rowoff;
    out[(((size_t)b * LDIM + l0 + row) * HDIM + h) * DDIM + e] = acc[r] / den[row];
  }
}

extern "C" void kernel_launch(void* const* d_in, const int* in_sizes, int n_in,
                              void* d_out, int out_size, void* d_ws, size_t ws_size,
                              hipStream_t stream) {
  const float* Q    = (const float*)d_in[0];
  const float* K    = (const float*)d_in[1];
  const float* V    = (const float*)d_in[2];
  const float* phiQ = (const float*)d_in[3];
  const float* phiK = (const float*)d_in[4];
  float* out = (float*)d_out;

  float* ws    = (float*)d_ws;
  float* qg    = ws;                                     // B*H*D
  float* ksum  = qg + BDIM * HDIM * DDIM;                // B*H*D
  float* kv    = ksum + BDIM * HDIM * DDIM;              // B*H*D*D
  float* alpha = kv + (size_t)BDIM * HDIM * DDIM * DDIM; // B*H*L

  int nz = BDIM * HDIM * DDIM * 2 + BDIM * HDIM * DDIM * DDIM;  // qg + ksum + kv (contiguous)
  rala_zero_kernel<<<(nz + 255) / 256, 256, 0, stream>>>(qg, nz);
  rala_qglobal_kernel<<<BDIM * HDIM * QSPLIT, 128, 0, stream>>>(Q, qg);
  rala_alpha_kernel<<<BDIM * HDIM, 256, 0, stream>>>(K, qg, alpha);
  rala_kv_kernel<<<BDIM * HDIM * LSPLIT, 256, 0, stream>>>(phiK, V, alpha, kv, ksum);
  rala_out_kernel<<<BDIM * HDIM * (LDIM / 16), 256, 0, stream>>>(phiQ, kv, ksum, out);
}